// Group_62603443306524
// MI455X (gfx1250) — compile-verified
//
#include <hip/hip_runtime.h>
#include <stdint.h>

// Problem constants (xyz: [8, 8192, 6] f32; outputs: neigh[8,512,32,3],
// neigh_n[8,512,32,3], center[8,512,3] concatenated flat in d_out).
#define BATCH 8
#define NPTS  8192
#define NGRP  512
#define GSZ   32
#define TILE  16      // centers per KNN workgroup (one WMMA row tile)
#define CHUNK 512     // points per KNN chunk (16 waves x 32 cols)
#define BGS3  (BATCH * NGRP * GSZ * 3)   // 393216 floats per [B,G,S,3] tensor

typedef __attribute__((ext_vector_type(2))) float v2f;
typedef __attribute__((ext_vector_type(8))) float v8f;

// Uniform-lane broadcasts: v_readlane_b32 (SGPR result) instead of the
// ds_bpermute_b32 + s_wait_dscnt round-trip that __shfl lowers to.
__device__ __forceinline__ float readlane_f(float v, int l) {
    return __int_as_float(__builtin_amdgcn_readlane(__float_as_int(v), l));
}
__device__ __forceinline__ int readlane_i(int v, int l) {
    return __builtin_amdgcn_readlane(v, l);
}

// ---------------------------------------------------------------------------
// Kernel 1: Farthest-point sampling. One workgroup per batch (8 blocks),
// 1024 threads (32 waves). Coordinates live in LDS (96KB, fits easily in the
// 320KB/WGP CDNA5 LDS); per-thread min-distances live in registers (8/thread).
// Writes the `center` section of d_out (also consumed by kernel 2).
// ---------------------------------------------------------------------------
__global__ void fps_kernel(const float* __restrict__ xyz, float* __restrict__ out)
{
    extern __shared__ char smem[];
    float* px = (float*)smem;           // [NPTS]
    float* py = px + NPTS;              // [NPTS]
    float* pz = py + NPTS;              // [NPTS]
    float* sD = pz + NPTS;              // [32] per-wave argmax value
    int*   sI = (int*)(sD + 32);        // [32] per-wave argmax index
    float* bc = (float*)(sI + 32);      // [4] broadcast new-center coords

    const int b    = blockIdx.x;
    const int tid  = threadIdx.x;       // 0..1023
    const int lane = tid & 31;
    const int wv   = tid >> 5;          // 0..31
    const float* base = xyz + (size_t)b * NPTS * 6;

    // Stage coords into LDS (coalesced AoS->SoA).
    for (int i = tid; i < NPTS; i += 1024) {
        px[i] = base[(size_t)i * 6 + 0];
        py[i] = base[(size_t)i * 6 + 1];
        pz[i] = base[(size_t)i * 6 + 2];
    }
    __syncthreads();

    // Blocked ownership: thread owns points [tid*8, tid*8+8). Coords + running
    // min-distance kept in registers; first center is point 0 (matches ref).
    float rx[8], ry[8], rz[8], rd[8];
    const float c0x = px[0], c0y = py[0], c0z = pz[0];
#pragma unroll
    for (int j = 0; j < 8; ++j) {
        const int p = tid * 8 + j;
        rx[j] = px[p]; ry[j] = py[p]; rz[j] = pz[p];
        const float dx = rx[j] - c0x, dy = ry[j] - c0y, dz = rz[j] - c0z;
        rd[j] = dx * dx + dy * dy + dz * dz;
    }

    float* centerOut = out + (size_t)2 * BGS3;   // center section of d_out
    if (tid == 0) {
        centerOut[((size_t)b * NGRP + 0) * 3 + 0] = c0x;
        centerOut[((size_t)b * NGRP + 0) * 3 + 1] = c0y;
        centerOut[((size_t)b * NGRP + 0) * 3 + 2] = c0z;
    }

    for (int k = 1; k < NGRP; ++k) {
        // Thread-local argmax (strict > keeps first index, like jnp.argmax).
        float md = rd[0];
        int   mi = tid * 8;
#pragma unroll
        for (int j = 1; j < 8; ++j)
            if (rd[j] > md) { md = rd[j]; mi = tid * 8 + j; }
        // Wave32 butterfly argmax reduction (tie -> lower index).
#pragma unroll
        for (int off = 16; off > 0; off >>= 1) {
            const float od = __shfl_xor(md, off);
            const int   oi = __shfl_xor(mi, off);
            if (od > md || (od == md && oi < mi)) { md = od; mi = oi; }
        }
        if (lane == 0) { sD[wv] = md; sI[wv] = mi; }
        __syncthreads();
        if (wv == 0) {
            float d = sD[lane];
            int   i = sI[lane];
#pragma unroll
            for (int off = 16; off > 0; off >>= 1) {
                const float od = __shfl_xor(d, off);
                const int   oi = __shfl_xor(i, off);
                if (od > d || (od == d && oi < i)) { d = od; i = oi; }
            }
            if (lane == 0) {
                const float cx = px[i], cy = py[i], cz = pz[i];
                bc[0] = cx; bc[1] = cy; bc[2] = cz;
                centerOut[((size_t)b * NGRP + k) * 3 + 0] = cx;
                centerOut[((size_t)b * NGRP + k) * 3 + 1] = cy;
                centerOut[((size_t)b * NGRP + k) * 3 + 2] = cz;
            }
        }
        __syncthreads();
        const float cx = bc[0], cy = bc[1], cz = bc[2];
#pragma unroll
        for (int j = 0; j < 8; ++j) {
            const float dx = rx[j] - cx, dy = ry[j] - cy, dz = rz[j] - cz;
            const float nd = dx * dx + dy * dy + dz * dz;
            rd[j] = fminf(rd[j], nd);
        }
        // No extra barrier needed: next iteration's sD/sI/bc writes are
        // ordered by the two barriers above.
    }
}

// ---------------------------------------------------------------------------
// Kernel 2: KNN + gather. Grid = (NGRP/TILE, BATCH), block = 512 (16 waves).
// d^2 = |c|^2 - 2 c.x + |x|^2 computed with V_WMMA_F32_16X16X4_F32:
//   A (16x4)  = [-2cx, -2cy, -2cz, 1]      per center row
//   B (4x16)  = [ x,    y,    z,  |x|^2]   per point column
//   C (16x16) = |c|^2 broadcast per row
// so one WMMA yields final d^2 for a 16-center x 16-point tile.
// A-layout per ISA: lanes 0-15 hold K=0,1 (vgpr0,vgpr1); lanes 16-31 K=2,3.
// B assumed mirrored (lanes 0-15: rows K=0,1; lanes 16-31: rows K=2,3).
// C/D: vgpr r -> row r (lanes 0-15) / row r+8 (lanes 16-31), col = lane&15.
// Each wave then owns one center and maintains a sorted top-32 (one entry
// per lane). Uniform broadcasts use v_readlane; only the positional shift
// in the insertion uses a real lane-permute.
// ---------------------------------------------------------------------------
__global__ void __launch_bounds__(512)
knn_kernel(const float* __restrict__ xyz, float* __restrict__ out)
{
    __shared__ float sx[CHUNK], sy[CHUNK], sz[CHUNK], sw[CHUNK]; // 8KB
    __shared__ float d2buf[TILE * CHUNK];                        // 32KB
    __shared__ float cen[TILE * 4];                              // x,y,z,|c|^2

    const int b        = blockIdx.y;
    const int tileBase = blockIdx.x * TILE;
    const int tid      = threadIdx.x;       // 0..511
    const int lane     = tid & 31;
    const int wv       = tid >> 5;          // 0..15
    const int hi       = (lane >> 4) & 1;
    const int l16      = lane & 15;
    const float* base  = xyz + (size_t)b * NPTS * 6;

    // Load this tile's centers from the center section of d_out (written by
    // the FPS kernel earlier on the same stream).
    if (tid < TILE) {
        const float* cp = out + (size_t)2 * BGS3
                        + ((size_t)b * NGRP + tileBase + tid) * 3;
        const float x = cp[0], y = cp[1], z = cp[2];
        cen[tid * 4 + 0] = x;
        cen[tid * 4 + 1] = y;
        cen[tid * 4 + 2] = z;
        cen[tid * 4 + 3] = x * x + y * y + z * z;
    }
    __syncthreads();

    // Build A (2 VGPRs) and C-init (8 VGPRs) once; reused every chunk.
    v2f A;
    {
        const float mx = cen[l16 * 4 + 0];
        const float my = cen[l16 * 4 + 1];
        const float mz = cen[l16 * 4 + 2];
        A[0] = hi ? (-2.0f * mz) : (-2.0f * mx);   // K=2 : K=0
        A[1] = hi ? 1.0f         : (-2.0f * my);   // K=3 : K=1
    }
    v8f Cinit;
#pragma unroll
    for (int r = 0; r < 8; ++r) Cinit[r] = cen[(r + 8 * hi) * 4 + 3];

    // Running top-32 for this wave's center: lane l holds l-th smallest.
    float topd = 3.4e38f;
    int   topi = 0;

#pragma unroll 1
    for (int chunk = 0; chunk < NPTS / CHUNK; ++chunk) {
        __syncthreads();   // previous selection done before ptsbuf overwrite
        {
            const int p = chunk * CHUNK + tid;
            const float x = base[(size_t)p * 6 + 0];
            const float y = base[(size_t)p * 6 + 1];
            const float z = base[(size_t)p * 6 + 2];
            sx[tid] = x; sy[tid] = y; sz[tid] = z;
            sw[tid] = x * x + y * y + z * z;
        }
        __syncthreads();

        // 2 WMMAs per wave -> 16 centers x 32 points of d^2 into LDS.
#pragma unroll
        for (int h = 0; h < 2; ++h) {
            const int col = wv * 32 + h * 16 + l16;
            v2f Bm;
            Bm[0] = hi ? sz[col] : sx[col];        // K=2 : K=0
            Bm[1] = hi ? sw[col] : sy[col];        // K=3 : K=1
            v8f acc = __builtin_amdgcn_wmma_f32_16x16x4_f32(
                false, A, false, Bm, (short)0, Cinit, false, false);
#pragma unroll
            for (int r = 0; r < 8; ++r)
                d2buf[(r + 8 * hi) * CHUNK + col] = acc[r];
        }
        __syncthreads();

        // Wave wv scans its center's 512 fresh distances.
        const float* row   = d2buf + wv * CHUNK;
        const int    gbase = chunk * CHUNK;
#pragma unroll 1
        for (int t = 0; t < CHUNK / 32; ++t) {
            const int   j  = t * 32 + lane;
            const float d  = row[j];
            const int   gi = gbase + j;
            const float kth = readlane_f(topd, 31);          // v_readlane
            unsigned int m = (unsigned int)__ballot(d < kth);
            while (m) {
                const int src = __builtin_ctz(m);            // uniform
                m &= m - 1;
                const float cd = readlane_f(d, src);         // v_readlane
                const int   ci = readlane_i(gi, src);        // v_readlane
                if (cd < readlane_f(topd, 31)) {             // scalar branch
                    const int pos =
                        __popc((unsigned int)__ballot(topd <= cd));
                    const float pd = __shfl_up(topd, 1);     // real permute
                    const int   pi = __shfl_up(topi, 1);
                    if (lane == pos)      { topd = cd; topi = ci; }
                    else if (lane > pos)  { topd = pd; topi = pi; }
                }
            }
        }
    }

    // Gather: wave wv -> center tileBase+wv, lane l -> l-th nearest neighbor.
    {
        const int   g  = tileBase + wv;
        const float cx = cen[wv * 4 + 0];
        const float cy = cen[wv * 4 + 1];
        const float cz = cen[wv * 4 + 2];
        const float* pp = base + (size_t)topi * 6;
        const float x  = pp[0], y  = pp[1], z  = pp[2];
        const float nx = pp[3], ny = pp[4], nz = pp[5];
        const size_t o  = (((size_t)b * NGRP + g) * GSZ + lane) * 3;
        out[o + 0] = x - cx;
        out[o + 1] = y - cy;
        out[o + 2] = z - cz;
        const size_t o2 = (size_t)BGS3 + o;
        out[o2 + 0] = nx;
        out[o2 + 1] = ny;
        out[o2 + 2] = nz;
    }
}

// ---------------------------------------------------------------------------
extern "C" void kernel_launch(void* const* d_in, const int* in_sizes, int n_in,
                              void* d_out, int out_size, void* d_ws, size_t ws_size,
                              hipStream_t stream)
{
    (void)in_sizes; (void)n_in; (void)d_ws; (void)ws_size; (void)out_size;
    const float* xyz = (const float*)d_in[0];
    float*       out = (float*)d_out;

    // FPS: one block per batch; dynamic LDS = 3*8192 f32 coords + reduction.
    const size_t smem1 = (size_t)(3 * NPTS) * sizeof(float)
                       + 32 * sizeof(float) + 32 * sizeof(int) + 4 * sizeof(float);
    fps_kernel<<<dim3(BATCH), dim3(1024), smem1, stream>>>(xyz, out);

    // KNN + gather: 32 center-tiles x 8 batches, 512 threads (16 waves).
    knn_kernel<<<dim3(NGRP / TILE, BATCH), dim3(512), 0, stream>>>(xyz, out);
}